// FusedConv_55465207660825
// MI455X (gfx1250) — compile-verified
//
#include <hip/hip_runtime.h>

typedef float v2f __attribute__((ext_vector_type(2)));
typedef float v8f __attribute__((ext_vector_type(8)));

#define TILE_E   32
#define EDIM     32
#define RNUM     1024
#define RHALF    512
#define RSTRIDE  516   // 512+4 pad: 8*516 % 64 == 32 -> writer half-waves hit disjoint banks
#define HSTRIDE  36    // 16B-aligned rows; 36*m % 64 = 4*(9m%16) -> 16 distinct banks
#define XSTRIDE  68    // 16B-aligned rows

__device__ __forceinline__ float silu_f(float x) {
  // x * sigmoid(x); v_rcp_f32 instead of IEEE divide
  return x * __builtin_amdgcn_rcpf(1.0f + __expf(-x));
}

__global__ __launch_bounds__(256) void fused_edge_kernel(
    const float* __restrict__ node_attr,
    const int*   __restrict__ edge_src,
    const int*   __restrict__ edge_dst,
    const float* __restrict__ edge_attr,
    const float* __restrict__ edge_sh,
    const float* __restrict__ w1,
    const float* __restrict__ b1,
    const float* __restrict__ w2,
    const float* __restrict__ b2,
    float* __restrict__ agg,
    float* __restrict__ deg,
    int n_edges)
{
  __shared__ __attribute__((aligned(16))) float sA[TILE_E * HSTRIDE];   // edge_attr [32][32]
  __shared__ __attribute__((aligned(16))) float sH[TILE_E * HSTRIDE];   // silu(hidden) [32][32]
  __shared__ __attribute__((aligned(16))) float sX[TILE_E * XSTRIDE];   // node_attr[src] [32][64]
  __shared__ __attribute__((aligned(16))) float sSh[TILE_E * 4];        // edge_sh
  __shared__ __attribute__((aligned(16))) float sR[TILE_E * RSTRIDE];   // radial half [32][512]

  const int t    = threadIdx.x;
  const int wid  = t >> 5;        // wave id (wave32)
  const int lane = t & 31;
  const int e0   = blockIdx.x * TILE_E;

  // ---------------- stage inputs into LDS ----------------
  {
    int le  = t >> 3;             // 0..31 edge within tile
    int sub = t & 7;              // 0..7
    int e   = e0 + le;
    bool ok = (e < n_edges);

    float4 ea = {0.f, 0.f, 0.f, 0.f};
    if (ok) ea = *(const float4*)(edge_attr + (size_t)e * EDIM + sub * 4);
    *(float4*)(&sA[le * HSTRIDE + sub * 4]) = ea;

    int s = ok ? edge_src[e] : 0;
    float4 xa = {0.f, 0.f, 0.f, 0.f}, xb = {0.f, 0.f, 0.f, 0.f};
    if (ok) {
      const float* np = node_attr + (size_t)s * 64 + sub * 8;
      xa = *(const float4*)(np);
      xb = *(const float4*)(np + 4);
    }
    *(float4*)(&sX[le * XSTRIDE + sub * 8])     = xa;
    *(float4*)(&sX[le * XSTRIDE + sub * 8 + 4]) = xb;

    if (t < TILE_E * 4) {
      int le2 = t >> 2, j = t & 3;
      int e2 = e0 + le2;
      sSh[t] = (e2 < n_edges) ? edge_sh[(size_t)e2 * 4 + j] : 0.0f;
    }
  }
  __syncthreads();

  const int m  = lane & 15;       // row/col within a 16x16 tile
  const int kh = lane >> 4;       // half-wave selector

  // ---------------- GEMM1: hidden = silu(edge_attr @ w1 + b1)  (waves 0..3) ----------------
  if (wid < 4) {
    int mt    = wid >> 1;                 // M tile (edges 0-15 / 16-31)
    int rbase = mt * 16;
    int ncol  = (wid & 1) * 16 + m;
    v2f af[8], bf[8];
    #pragma unroll
    for (int k0 = 0; k0 < 8; ++k0) {
      int kb = 4 * k0 + 2 * kh;
      af[k0].x = sA[(rbase + m) * HSTRIDE + kb];
      af[k0].y = sA[(rbase + m) * HSTRIDE + kb + 1];
      bf[k0].x = w1[(kb + 0) * EDIM + ncol];
      bf[k0].y = w1[(kb + 1) * EDIM + ncol];
    }
    v8f acc = {};
    #pragma unroll
    for (int k0 = 0; k0 < 8; ++k0) {
      acc = __builtin_amdgcn_wmma_f32_16x16x4_f32(false, af[k0], false, bf[k0],
                                                  (short)0, acc, false, false);
    }
    float bias = b1[ncol];
    #pragma unroll
    for (int j = 0; j < 8; ++j) {
      int row = rbase + j + 8 * kh;
      sH[row * HSTRIDE + ncol] = silu_f(acc[j] + bias);
    }
  }
  __syncthreads();

  // A-fragments for both M tiles, shared across all N tiles of this wave.
  v2f afA[8], afB[8];
  #pragma unroll
  for (int k0 = 0; k0 < 8; ++k0) {
    int kb = 4 * k0 + 2 * kh;
    afA[k0].x = sH[m * HSTRIDE + kb];
    afA[k0].y = sH[m * HSTRIDE + kb + 1];
    afB[k0].x = sH[(16 + m) * HSTRIDE + kb];
    afB[k0].y = sH[(16 + m) * HSTRIDE + kb + 1];
  }

  // Per-thread tensor-product state: thread t -> (edges le0 and le0+16, out col w)
  const int le0 = t >> 4;
  const int w   = t & 15;
  float s0[2], sx[2], sy[2], sz[2];
  #pragma unroll
  for (int ei = 0; ei < 2; ++ei) {
    int le = le0 + 16 * ei;
    s0[ei] = sSh[le * 4 + 0];
    sx[ei] = sSh[le * 4 + 1];
    sy[ei] = sSh[le * 4 + 2];
    sz[ei] = sSh[le * 4 + 3];
  }
  const float invs3 = 0.57735026918962576f;   // 1/sqrt(3)
  float o0[2] = {0.f, 0.f}, p01[2] = {0.f, 0.f};
  float q0[2] = {0.f, 0.f}, q1[2] = {0.f, 0.f}, q2[2] = {0.f, 0.f};

  // ---------------- GEMM2 + TP, two 512-column halves ----------------
  for (int half = 0; half < 2; ++half) {
    for (int nt = wid; nt < RHALF / 16; nt += 8) {
      int lcol = nt * 16 + m;
      int gcol = half * RHALF + lcol;
      v2f bf[8];
      #pragma unroll
      for (int k0 = 0; k0 < 8; ++k0) {
        int kb = 4 * k0 + 2 * kh;
        bf[k0].x = w2[(size_t)(kb + 0) * RNUM + gcol];
        bf[k0].y = w2[(size_t)(kb + 1) * RNUM + gcol];
      }
      v8f acc0 = {}, acc1 = {};
      #pragma unroll
      for (int k0 = 0; k0 < 8; ++k0) {   // two independent chains interleaved
        acc0 = __builtin_amdgcn_wmma_f32_16x16x4_f32(false, afA[k0], false, bf[k0],
                                                     (short)0, acc0, false, false);
        acc1 = __builtin_amdgcn_wmma_f32_16x16x4_f32(false, afB[k0], false, bf[k0],
                                                     (short)0, acc1, false, false);
      }
      float bias = b2[gcol];
      #pragma unroll
      for (int j = 0; j < 8; ++j) {
        sR[(j + 8 * kh) * RSTRIDE + lcol]      = acc0[j] + bias;
        sR[(16 + j + 8 * kh) * RSTRIDE + lcol] = acc1[j] + bias;
      }
    }
    __syncthreads();

    // tensor-product partial over this half, for both edges of this thread
    #pragma unroll
    for (int ei = 0; ei < 2; ++ei) {
      int le = le0 + 16 * ei;
      const float* rr = &sR[le * RSTRIDE];
      const float* xr = &sX[le * XSTRIDE];
      if (half == 0) {
        // paths: w00 (local 0..255), w01 (local 256..511)
        #pragma unroll
        for (int u = 0; u < 16; ++u) {
          float x0u  = xr[u];
          float w00v = rr[u * 16 + w];
          float w01v = rr[256 + u * 16 + w];
          o0[ei]  = fmaf(x0u * s0[ei], w00v, o0[ei]);
          p01[ei] = fmaf(x0u, w01v, p01[ei]);
        }
      } else {
        // paths: w10 (local 0..255), w11 (local 256..511)
        #pragma unroll
        for (int u = 0; u < 16; ++u) {
          float x1x  = xr[16 + u * 3 + 0];
          float x1y  = xr[16 + u * 3 + 1];
          float x1z  = xr[16 + u * 3 + 2];
          float w10v = rr[u * 16 + w];
          float w11v = rr[256 + u * 16 + w];
          float a1   = x1x * sx[ei] + x1y * sy[ei] + x1z * sz[ei];
          o0[ei] = fmaf(a1 * invs3, w11v, o0[ei]);
          q0[ei] = fmaf(x1x, w10v, q0[ei]);
          q1[ei] = fmaf(x1y, w10v, q1[ei]);
          q2[ei] = fmaf(x1z, w10v, q2[ei]);
        }
      }
    }
    __syncthreads();
  }

  // ---------------- scatter ----------------
  {
    const float alpha = 0.17677669529663687f;  // 1/sqrt(2*MUL)
    const float c1    = alpha * invs3;
    #pragma unroll
    for (int ei = 0; ei < 2; ++ei) {
      int e = e0 + le0 + 16 * ei;
      if (e < n_edges) {
        float out0 = alpha * o0[ei];
        float o1x  = c1 * (sx[ei] * p01[ei] + s0[ei] * q0[ei]);
        float o1y  = c1 * (sy[ei] * p01[ei] + s0[ei] * q1[ei]);
        float o1z  = c1 * (sz[ei] * p01[ei] + s0[ei] * q2[ei]);
        int d = edge_dst[e];
        float* ap = agg + (size_t)d * 64;
        atomicAdd(&ap[w], out0);
        atomicAdd(&ap[16 + w * 3 + 0], o1x);
        atomicAdd(&ap[16 + w * 3 + 1], o1y);
        atomicAdd(&ap[16 + w * 3 + 2], o1z);
        if (w == 0) atomicAdd(&deg[edge_src[e]], 1.0f);  // reference: degree over src
      }
    }
  }
}

__global__ void finalize_kernel(float* __restrict__ out,
                                const float* __restrict__ deg, int total) {
  int i = blockIdx.x * blockDim.x + threadIdx.x;
  if (i >= total) return;
  float d = deg[i >> 6];
  out[i] = out[i] / fmaxf(d, 1.0f);
}

extern "C" void kernel_launch(void* const* d_in, const int* in_sizes, int n_in,
                              void* d_out, int out_size, void* d_ws, size_t ws_size,
                              hipStream_t stream) {
  const float* node_attr = (const float*)d_in[0];
  const int*   edge_idx  = (const int*)  d_in[1];
  const float* edge_attr = (const float*)d_in[2];
  const float* edge_sh   = (const float*)d_in[3];
  const float* w1        = (const float*)d_in[4];
  const float* b1        = (const float*)d_in[5];
  const float* w2        = (const float*)d_in[6];
  const float* b2        = (const float*)d_in[7];

  int n_edges = in_sizes[2] / EDIM;
  int n_nodes = in_sizes[0] / 64;

  float* out = (float*)d_out;
  float* deg = (float*)d_ws;

  hipMemsetAsync(out, 0, (size_t)n_nodes * 64 * sizeof(float), stream);
  hipMemsetAsync(deg, 0, (size_t)n_nodes * sizeof(float), stream);

  const int* src = edge_idx;
  const int* dst = edge_idx + n_edges;

  int nblk = (n_edges + TILE_E - 1) / TILE_E;
  hipLaunchKernelGGL(fused_edge_kernel, dim3(nblk), dim3(256), 0, stream,
                     node_attr, src, dst, edge_attr, edge_sh,
                     w1, b1, w2, b2, out, deg, n_edges);

  int total = n_nodes * 64;
  hipLaunchKernelGGL(finalize_kernel, dim3((total + 255) / 256), dim3(256), 0, stream,
                     out, deg, total);
}